// GravNetModel_7473243095611
// MI455X (gfx1250) — compile-verified
//
#include <hip/hip_runtime.h>
#include <hip/hip_bf16.h>
#include <math.h>

// ---------------------------------------------------------------------------
// GravNet model for MI455X (gfx1250, wave32, WMMA).
// Dense 64-wide GEMMs use native fp32 WMMA: V_WMMA_F32_16X16X4_F32,
// chained 16x over K=64 per 16x16 output tile. kNN is LDS-tiled with
// register-resident top-k. Everything fits in L2 (192MB) -> latency bound.
// ---------------------------------------------------------------------------

typedef __attribute__((ext_vector_type(2))) float v2f;
typedef __attribute__((ext_vector_type(8))) float v8f;

#define NN      12288
#define HID     64
#define NG      8
#define LN_EPS  1e-5f

// ------------------------- input linear: [N,3] @ [3,64] --------------------
__global__ void input_linear_kernel(const float* __restrict__ x,
                                    const float* __restrict__ W,   // [3,64]
                                    const float* __restrict__ b,   // [64]
                                    float* __restrict__ h)         // [N,64]
{
    int idx = blockIdx.x * blockDim.x + threadIdx.x;     // N*64
    if (idx >= NN * HID) return;
    int i = idx >> 6, j = idx & 63;
    const float* xr = x + i * 3;
    h[idx] = b[j] + xr[0] * W[j] + xr[1] * W[64 + j] + xr[2] * W[128 + j];
}

// ------------------- small linear: [N,64] @ [64,F] + bias ------------------
__global__ void linear_small_kernel(const float* __restrict__ in,  // [N,64]
                                    const float* __restrict__ W,   // [64,F]
                                    const float* __restrict__ b,   // [F]
                                    float* __restrict__ out,       // [N,F]
                                    int F)
{
    int idx = blockIdx.x * blockDim.x + threadIdx.x;     // N*F
    if (idx >= NN * F) return;
    int i = idx / F, f = idx % F;
    const float* r = in + i * HID;
    float acc = b[f];
#pragma unroll 8
    for (int j = 0; j < HID; ++j) acc += r[j] * W[j * F + f];
    out[idx] = acc;
}

// -------------------------- kNN in 4-D coord space -------------------------
// One thread per query node; candidates staged through LDS in tiles of 256.
// Register-resident sorted top-k (ascending by squared distance).
template <int K>
__global__ void knn_kernel(const float* __restrict__ s,    // [N,4] (float4)
                           int* __restrict__ knn_idx,      // [N,K]
                           float* __restrict__ knn_w)      // [N,K]
{
    __shared__ float4 tile[256];
    const int i = blockIdx.x * 256 + threadIdx.x;
    const float4* s4 = reinterpret_cast<const float4*>(s);
    const float4 si = s4[i];
    const float sqi = si.x * si.x + si.y * si.y + si.z * si.z + si.w * si.w;

    float bd[K];
    int   bi[K];
#pragma unroll
    for (int t = 0; t < K; ++t) { bd[t] = INFINITY; bi[t] = 0; }

    for (int base = 0; base < NN; base += 256) {
        __syncthreads();
        tile[threadIdx.x] = s4[base + threadIdx.x];
        if (base + 256 < NN)  // gfx1250: global_prefetch_b8 of next tile
            __builtin_prefetch(&s4[base + 256 + threadIdx.x], 0, 0);
        __syncthreads();
        for (int t = 0; t < 256; ++t) {
            float4 sj = tile[t];
            float sqj = sj.x * sj.x + sj.y * sj.y + sj.z * sj.z + sj.w * sj.w;
            float dot = si.x * sj.x + si.y * sj.y + si.z * sj.z + si.w * sj.w;
            float d = sqi + sqj - 2.0f * dot;           // matches reference formula
            if (d < bd[K - 1]) {
                float cd = d; int ci = base + t;
#pragma unroll
                for (int u = 0; u < K; ++u) {
                    if (cd < bd[u]) {
                        float td = bd[u]; int ti = bi[u];
                        bd[u] = cd; bi[u] = ci;
                        cd = td;    ci = ti;
                    }
                }
            }
        }
    }
#pragma unroll
    for (int t = 0; t < K; ++t) {
        knn_idx[i * K + t] = bi[t];
        knn_w[i * K + t]   = expf(-10.0f * fmaxf(bd[t], 0.0f));
    }
}

// --------------------- weighted mean + max aggregation ---------------------
__global__ void aggregate_kernel(const float* __restrict__ hp,      // [N,32]
                                 const int* __restrict__ knn_idx,   // [N,k]
                                 const float* __restrict__ knn_w,   // [N,k]
                                 int k,
                                 float* __restrict__ agg)           // [N,64]
{
    int idx = blockIdx.x * blockDim.x + threadIdx.x;    // N*32
    if (idx >= NN * 32) return;
    int i = idx >> 5, p = idx & 31;
    float sum = 0.f, mx = -INFINITY;
    for (int t = 0; t < k; ++t) {
        int   j = knn_idx[i * k + t];
        float w = knn_w[i * k + t];
        float v = hp[j * 32 + p] * w;
        sum += v;
        mx = fmaxf(mx, v);
    }
    agg[i * 64 + p]      = sum / (float)k;
    agg[i * 64 + 32 + p] = mx;
}

// ------------- WMMA fp32 GEMM: out = A@W1 (+ B@W2) + bias, 64-wide ---------
// 8 waves/block; each wave owns one 16x16 output tile (2 row-tiles x 4
// col-tiles = 32 rows x 64 cols per block). K=64 consumed in 16 chained
// V_WMMA_F32_16X16X4_F32 ops. EXEC is all-ones (exact grid, uniform branches).
//
// Wave32 fp32 WMMA layout:
//   A (16x4): lane l holds A[l%16][k0], A[l%16][k0+1], k0 = (l<16)?0:2
//   B (4x16): lane l holds B[k0][l%16], B[k0+1][l%16]
//   C/D (16x16): VGPR v: lanes 0-15 -> (M=v, N=l); lanes 16-31 -> (M=v+8, N=l-16)
__global__ void gemm64_wmma_kernel(const float* __restrict__ A,    // [N,64]
                                   const float* __restrict__ W1,   // [64,64]
                                   const float* __restrict__ B,    // [N,64] or null
                                   const float* __restrict__ W2,   // [64,64] or null
                                   const float* __restrict__ bias, // [64]
                                   float* __restrict__ out)        // [N,64]
{
    const int lane = threadIdx.x & 31;
    const int wave = threadIdx.x >> 5;
    const int trow = blockIdx.x * 2 + (wave >> 2);   // 16-row tile index
    const int tcol = wave & 3;                       // 16-col tile index
    const int mrow = trow * 16 + (lane & 15);
    const int ncol = tcol * 16 + (lane & 15);
    const int koff = (lane < 16) ? 0 : 2;

    v8f acc = {0.f, 0.f, 0.f, 0.f, 0.f, 0.f, 0.f, 0.f};

    const float* arow = A + mrow * HID;
#pragma unroll
    for (int kb = 0; kb < HID; kb += 4) {
        v2f a; a.x = arow[kb + koff];            a.y = arow[kb + koff + 1];
        v2f b; b.x = W1[(kb + koff) * HID + ncol]; b.y = W1[(kb + koff + 1) * HID + ncol];
        acc = __builtin_amdgcn_wmma_f32_16x16x4_f32(false, a, false, b,
                                                    (short)0, acc, false, false);
    }
    if (B != nullptr) {                          // uniform branch: EXEC unchanged
        const float* brow = B + mrow * HID;
#pragma unroll
        for (int kb = 0; kb < HID; kb += 4) {
            v2f a; a.x = brow[kb + koff];            a.y = brow[kb + koff + 1];
            v2f b; b.x = W2[(kb + koff) * HID + ncol]; b.y = W2[(kb + koff + 1) * HID + ncol];
            acc = __builtin_amdgcn_wmma_f32_16x16x4_f32(false, a, false, b,
                                                        (short)0, acc, false, false);
        }
    }

    const float bj = bias[ncol];
    const int rbase = trow * 16 + ((lane < 16) ? 0 : 8);
#pragma unroll
    for (int v = 0; v < 8; ++v)
        out[(rbase + v) * HID + ncol] = acc[v] + bj;
}

// ----------------- LayerNorm(64) + ReLU + residual (in-place h) ------------
__global__ void ln_relu_res_kernel(const float* __restrict__ y,     // [N,64]
                                   const float* __restrict__ gamma, // [64]
                                   const float* __restrict__ beta,  // [64]
                                   float* __restrict__ h)           // [N,64] in/out
{
    int i = blockIdx.x * blockDim.x + threadIdx.x;
    if (i >= NN) return;
    const float* r = y + i * HID;
    float m = 0.f;
#pragma unroll 8
    for (int j = 0; j < HID; ++j) m += r[j];
    m *= (1.f / 64.f);
    float v = 0.f;
#pragma unroll 8
    for (int j = 0; j < HID; ++j) { float d = r[j] - m; v += d * d; }
    v *= (1.f / 64.f);
    float inv = rsqrtf(v + LN_EPS);
#pragma unroll 8
    for (int j = 0; j < HID; ++j) {
        float val = (r[j] - m) * inv * gamma[j] + beta[j];
        h[i * HID + j] = fmaxf(val, 0.f) + h[i * HID + j];
    }
}

// ------------------ deterministic segment-mean pooling ---------------------
__global__ void pool_kernel(const float* __restrict__ h,     // [N,64]
                            const int* __restrict__ batch,   // [N]
                            float* __restrict__ pooled)      // [8,64]
{
    int g = blockIdx.x, j = threadIdx.x;
    float acc = 0.f; int cnt = 0;
    for (int i = 0; i < NN; ++i) {
        if (batch[i] == g) { acc += h[i * HID + j]; cnt++; }
    }
    pooled[g * HID + j] = acc / fmaxf((float)cnt, 1.0f);
}

// ----------------------------- MLP head ------------------------------------
__global__ void head_kernel(const float* __restrict__ pooled, // [8,64]
                            const float* __restrict__ wm,     // [64,128]
                            const float* __restrict__ bm,     // [128]
                            const float* __restrict__ mg,     // [128]
                            const float* __restrict__ mbn,    // [128]
                            const float* __restrict__ wc,     // [128,4]
                            const float* __restrict__ bc,     // [4]
                            const float* __restrict__ we,     // [128,1]
                            const float* __restrict__ be,     // [1]
                            float* __restrict__ out)          // [40]: logits[32], energy[8]
{
    __shared__ float e[128];
    int g = blockIdx.x, t = threadIdx.x;
    float acc = bm[t];
    for (int j = 0; j < HID; ++j) acc += pooled[g * HID + j] * wm[j * 128 + t];
    acc = fmaxf(acc, 0.f);                 // relu BEFORE layernorm (matches ref)
    e[t] = acc;
    __syncthreads();
    float m = 0.f;
    for (int j = 0; j < 128; ++j) m += e[j];
    m *= (1.f / 128.f);
    float v = 0.f;
    for (int j = 0; j < 128; ++j) { float d = e[j] - m; v += d * d; }
    v *= (1.f / 128.f);
    float en = (acc - m) * rsqrtf(v + LN_EPS) * mg[t] + mbn[t];
    __syncthreads();
    e[t] = en;
    __syncthreads();
    if (t < 4) {
        float o = bc[t];
        for (int j = 0; j < 128; ++j) o += e[j] * wc[j * 4 + t];
        out[g * 4 + t] = o;
    } else if (t == 4) {
        float o = be[0];
        for (int j = 0; j < 128; ++j) o += e[j] * we[j];
        out[32 + g] = o;
    }
}

// ---------------------------------------------------------------------------
// Input leaf order (JAX pytree: dict keys sorted at every level):
//   0 batch[12288]  1 edge_index[2*65536] (unused)
//   2..34 blocks b=0..2, each 11 leaves sorted:
//     +0 bh[32] +1 bl[64] +2 bn[64] +3 bo2[64] +4 bs[4] +5 g[64]
//     +6 wh[64*32] +7 wl[64*64] +8 wo1[64*64] +9 wo2[64*64] +10 ws[64*4]
//   35 cls.b[4] 36 cls.w[128*4] 37 energy.b[1] 38 energy.w[128]
//   39 input.b[64] 40 input.w[3*64]
//   41 mlp.b[128] 42 mlp.bn[128] 43 mlp.g[128] 44 mlp.w[64*128]
//   45 x[12288*3]
// ---------------------------------------------------------------------------
extern "C" void kernel_launch(void* const* d_in, const int* in_sizes, int n_in,
                              void* d_out, int out_size, void* d_ws, size_t ws_size,
                              hipStream_t stream)
{
    (void)in_sizes; (void)n_in; (void)out_size; (void)ws_size;

    auto F = [&](int i) { return (const float*)d_in[i]; };
    const int* batch = (const int*)d_in[0];
    const float* x      = F(45);
    const float* in_b   = F(39);
    const float* in_w   = F(40);
    const float* mlp_b  = F(41);
    const float* mlp_bn = F(42);
    const float* mlp_g  = F(43);
    const float* mlp_w  = F(44);
    const float* cls_b  = F(35);
    const float* cls_w  = F(36);
    const float* en_b   = F(37);
    const float* en_w   = F(38);
    float* out = (float*)d_out;

    // workspace carve-out (256-byte aligned slabs)
    char* ws = (char*)d_ws;
    size_t off = 0;
    auto carve = [&](size_t bytes) -> void* {
        void* p = ws + off;
        off = (off + bytes + 255) & ~(size_t)255;
        return p;
    };
    float* h      = (float*)carve((size_t)NN * HID * 4);
    float* y1     = (float*)carve((size_t)NN * HID * 4);
    float* y2     = (float*)carve((size_t)NN * HID * 4);
    float* agg    = (float*)carve((size_t)NN * HID * 4);
    float* sc     = (float*)carve((size_t)NN * 4 * 4);
    float* hp     = (float*)carve((size_t)NN * 32 * 4);
    float* knnw   = (float*)carve((size_t)NN * 32 * 4);
    int*   knni   = (int*)  carve((size_t)NN * 32 * 4);
    float* pooled = (float*)carve((size_t)NG * HID * 4);

    // 1) input linear
    input_linear_kernel<<<(NN * HID) / 256, 256, 0, stream>>>(x, in_w, in_b, h);

    // 2) three GravNet blocks
    const int ks_arr[3] = {32, 16, 16};
    for (int b = 0; b < 3; ++b) {
        const int base = 2 + 11 * b;
        const float* bh  = F(base + 0);
        const float* bl  = F(base + 1);
        const float* bn  = F(base + 2);
        const float* bo2 = F(base + 3);
        const float* bs  = F(base + 4);
        const float* gg  = F(base + 5);
        const float* wh  = F(base + 6);
        const float* wl  = F(base + 7);
        const float* wo1 = F(base + 8);
        const float* wo2 = F(base + 9);
        const float* wsp = F(base + 10);
        const int k = ks_arr[b];

        linear_small_kernel<<<(NN * 4) / 256, 256, 0, stream>>>(h, wsp, bs, sc, 4);
        linear_small_kernel<<<(NN * 32) / 256, 256, 0, stream>>>(h, wh, bh, hp, 32);

        if (k == 32)
            knn_kernel<32><<<NN / 256, 256, 0, stream>>>(sc, knni, knnw);
        else
            knn_kernel<16><<<NN / 256, 256, 0, stream>>>(sc, knni, knnw);

        aggregate_kernel<<<(NN * 32) / 256, 256, 0, stream>>>(hp, knni, knnw, k, agg);

        // y1 = h@wo1 + agg@wo2 + bo2   (dual WMMA GEMM)
        gemm64_wmma_kernel<<<NN / 32, 256, 0, stream>>>(h, wo1, agg, wo2, bo2, y1);
        // y2 = y1@wl + bl              (single WMMA GEMM)
        gemm64_wmma_kernel<<<NN / 32, 256, 0, stream>>>(y1, wl, nullptr, nullptr, bl, y2);
        // h = relu(LN(y2)*g+bn) + h
        ln_relu_res_kernel<<<(NN + 255) / 256, 256, 0, stream>>>(y2, gg, bn, h);
    }

    // 3) segment-mean pool + head
    pool_kernel<<<NG, HID, 0, stream>>>(h, batch, pooled);
    head_kernel<<<NG, 128, 0, stream>>>(pooled, mlp_w, mlp_b, mlp_g, mlp_bn,
                                        cls_w, cls_b, en_w, en_b, out);
}